// GNN_85856396247479
// MI455X (gfx1250) — compile-verified
//
#include <hip/hip_runtime.h>

// ---------------------------------------------------------------------------
// GraphSAGE 2-layer inference for MI455X (gfx1250, wave32, WMMA).
//   layer: out = relu?( (scatter_mean(x, edges)) @ W_l + b + x @ W_r )
// Memory-bound on edge traffic (~1 GB @ 23.3 TB/s); GEMMs (~5 GFLOP) run on
// the native f32 WMMA path (V_WMMA_F32_16X16X4_F32) to keep full precision.
// ---------------------------------------------------------------------------

typedef __attribute__((ext_vector_type(2))) float v2f;
typedef __attribute__((ext_vector_type(8))) float v8f;

#ifndef SAGE_IN_C
#define SAGE_IN_C 128
#endif

// ----------------------------- zero fill -----------------------------------
__global__ void sage_zero_kernel(float* __restrict__ p, int n) {
  int i = blockIdx.x * blockDim.x + threadIdx.x;
  if (i < n) p[i] = 0.0f;
}

// ------------------------ edge scatter-add (mean num.) ---------------------
// One wave32 per edge; 32 lanes sweep the C channels (C/32 coalesced passes).
// agg[dst] += feat[src]; optionally cnt[dst] += 1 (degree, layer 1 only).
// agg (25.6 MB) lives in the 192 MB L2, so atomics resolve on-chip.
template <int C>
__global__ void sage_scatter_kernel(const float* __restrict__ feat,
                                    const long long* __restrict__ src,
                                    const long long* __restrict__ dst,
                                    float* __restrict__ agg,
                                    float* __restrict__ cnt,
                                    int nEdges, int addCount) {
  int wid  = (blockIdx.x * blockDim.x + threadIdx.x) >> 5;
  int lane = threadIdx.x & 31;
  if (wid >= nEdges) return;

  long long s = src[wid];
  long long d = dst[wid];
  const float* fs = feat + s * (long long)C;
  float*       ad = agg  + d * (long long)C;

  if (addCount && lane == 0) atomicAdd(cnt + d, 1.0f);

#pragma unroll
  for (int c0 = 0; c0 < C; c0 += 32) {
    atomicAdd(ad + c0 + lane, fs[c0 + lane]);
  }
}

// ------------------- fused SAGE GEMM epilogue (WMMA f32) -------------------
// One wave computes a 16x16 tile of:  Out = act( (Agg/cnt)@Wl + bias + X@Wr )
// K is a multiple of 4; two V_WMMA_F32_16X16X4_F32 per K-step (mean + root).
//
// Fragment layouts (CDNA5 ISA 7.12.2, 32-bit, wave32):
//   A 16x4 : lane&15 = row M;  (lane>>4)*2 = K offset; 2 VGPRs hold K,K+1
//   B 4x16 : lane&15 = col N;  (lane>>4)*2 = K offset; 2 VGPRs hold K,K+1
//   C/D    : reg r, lane l -> row = r + 8*(l>>4), col = l&15
template <int K, int DO_RELU>
__global__ void sage_gemm_kernel(const float* __restrict__ Agg,   // [M x K]
                                 const float* __restrict__ cnt,   // [M]
                                 const float* __restrict__ X,     // [M x K]
                                 const float* __restrict__ Wl,    // [K x N]
                                 const float* __restrict__ Wr,    // [K x N]
                                 const float* __restrict__ bias,  // [N]
                                 float* __restrict__ Out,         // [M x N]
                                 int M, int N) {
  int wid  = (blockIdx.x * blockDim.x + threadIdx.x) >> 5;  // wave id
  int lane = threadIdx.x & 31;

  int nTiles = N >> 4;
  int mTile  = wid / nTiles;
  int nTile  = wid - mTile * nTiles;
  if (mTile * 16 >= M) return;  // wave-uniform: EXEC stays all-ones for WMMA

  const int m0 = mTile * 16;
  const int n0 = nTile * 16;

  const int rowA  = m0 + (lane & 15);        // A-fragment row for this lane
  const int colB  = n0 + (lane & 15);        // B-fragment col for this lane
  const int kHalf = (lane >> 4) << 1;        // 0 or 2 within each K=4 slab

  // mean divisor fused into the A load
  const float invc = 1.0f / fmaxf(cnt[rowA], 1.0f);

  // bias folded into the accumulator (every C/D reg of a lane = same column)
  const float bval = bias[colB];
  v8f c;
#pragma unroll
  for (int i = 0; i < 8; ++i) c[i] = bval;

  const float* aRowM = Agg + (long long)rowA * K;
  const float* aRowX = X   + (long long)rowA * K;

#pragma unroll 4
  for (int k = 0; k < K; k += 4) {
    const int ka = k + kHalf;

    v2f am = *(const v2f*)(aRowM + ka);  // 8B aligned: ka even, row*K*4 | 512
    am = am * invc;
    v2f ax = *(const v2f*)(aRowX + ka);

    v2f bl;
    bl.x = Wl[(long long)ka * N + colB];
    bl.y = Wl[(long long)(ka + 1) * N + colB];
    v2f br;
    br.x = Wr[(long long)ka * N + colB];
    br.y = Wr[(long long)(ka + 1) * N + colB];

    // (neg_a, A, neg_b, B, c_mod, C, reuse_a, reuse_b)
    c = __builtin_amdgcn_wmma_f32_16x16x4_f32(false, am, false, bl,
                                              (short)0, c, false, false);
    c = __builtin_amdgcn_wmma_f32_16x16x4_f32(false, ax, false, br,
                                              (short)0, c, false, false);
  }

  const int rowBase = m0 + ((lane >> 4) << 3);  // +8 for upper lane half
  const int col     = n0 + (lane & 15);
#pragma unroll
  for (int r = 0; r < 8; ++r) {
    float v = c[r];
    if (DO_RELU) v = fmaxf(v, 0.0f);
    Out[(long long)(rowBase + r) * N + col] = v;
  }
}

// ------------------------------ host launch --------------------------------
extern "C" void kernel_launch(void* const* d_in, const int* in_sizes, int n_in,
                              void* d_out, int out_size, void* d_ws,
                              size_t ws_size, hipStream_t stream) {
  const float*     x   = (const float*)d_in[0];
  const long long* ei  = (const long long*)d_in[1];  // int64 [2, E]
  const float*     W1l = (const float*)d_in[2];
  const float*     b1  = (const float*)d_in[3];
  const float*     W1r = (const float*)d_in[4];
  const float*     W2l = (const float*)d_in[5];
  const float*     b2  = (const float*)d_in[6];
  const float*     W2r = (const float*)d_in[7];
  float*           out = (float*)d_out;

  const int IN_C = 128, HID_C = 128, OUT_C = 64;
  const int nNodes = in_sizes[0] / IN_C;   // 50000
  const int nEdges = in_sizes[1] / 2;      // 800000
  const long long* src = ei;
  const long long* dst = ei + nEdges;

  // workspace: [cnt (nNodes)] [agg (nNodes*128)] [h (nNodes*128)]
  float* cnt = (float*)d_ws;
  float* agg = cnt + nNodes;
  float* h   = agg + (long long)nNodes * HID_C;

  const int BLK = 256;  // 8 wave32s per block

  // ---- layer 1 ----
  {
    int nz = nNodes + nNodes * IN_C;  // cnt + agg are adjacent
    sage_zero_kernel<<<(nz + BLK - 1) / BLK, BLK, 0, stream>>>(cnt, nz);
  }
  {
    int waves  = nEdges;
    int blocks = (waves * 32 + BLK - 1) / BLK;
    sage_scatter_kernel<128><<<blocks, BLK, 0, stream>>>(
        x, src, dst, agg, cnt, nEdges, /*addCount=*/1);
  }
  {
    int waves  = (nNodes / 16) * (HID_C / 16);  // 3125 * 8
    int blocks = (waves * 32 + BLK - 1) / BLK;
    sage_gemm_kernel<128, 1><<<blocks, BLK, 0, stream>>>(
        agg, cnt, x, W1l, W1r, b1, h, nNodes, HID_C);
  }

  // ---- layer 2 (dropout is identity at inference) ----
  {
    int nz = nNodes * HID_C;
    sage_zero_kernel<<<(nz + BLK - 1) / BLK, BLK, 0, stream>>>(agg, nz);
  }
  {
    int waves  = nEdges;
    int blocks = (waves * 32 + BLK - 1) / BLK;
    sage_scatter_kernel<128><<<blocks, BLK, 0, stream>>>(
        h, src, dst, agg, cnt, nEdges, /*addCount=*/0);
  }
  {
    int waves  = (nNodes / 16) * (OUT_C / 16);  // 3125 * 4
    int blocks = (waves * 32 + BLK - 1) / BLK;
    sage_gemm_kernel<128, 0><<<blocks, BLK, 0, stream>>>(
        agg, cnt, h, W2l, W2r, b2, out, nNodes, OUT_C);
  }
}